// DRMM_1503238554328
// MI455X (gfx1250) — compile-verified
//
#include <hip/hip_runtime.h>
#include <hip/hip_bf16.h>
#include <math.h>

typedef __attribute__((ext_vector_type(2))) float v2f;
typedef __attribute__((ext_vector_type(8))) float v8f;

#define BB   64
#define LQ   16
#define LD   1024
#define EDIM 300
#define BINS 30
#define DT   32          // doc rows per block (2 waves x 16)
#define LPAD 301         // odd stride -> conflict-free LDS rows

// ---------------------------------------------------------------------------
// Kernel 0: zero the global histogram workspace (ws is poisoned to 0xAA).
// ---------------------------------------------------------------------------
__global__ void drmm_zero_hist(int* __restrict__ ghist, int n) {
    int i = blockIdx.x * blockDim.x + threadIdx.x;
    if (i < n) ghist[i] = 0;
}

// ---------------------------------------------------------------------------
// Kernel 1: cosine tiles via v_wmma_f32_16x16x4_f32 + per-(b,q) histogram.
// Grid: (LD/DT, B). Block: 64 threads = 2 waves; wave w owns doc rows
// [w*16, w*16+16) of this block's 32-doc tile.
// ---------------------------------------------------------------------------
__global__ __launch_bounds__(64) void drmm_cos_hist(
    const int*   __restrict__ qtok,
    const int*   __restrict__ dtok,
    const float* __restrict__ emb,
    int*         __restrict__ ghist)   // [B][LQ][BINS] ints
{
    __shared__ float qs[LQ][LPAD];
    __shared__ float ds[DT][LPAD];
    __shared__ float qinv[LQ];
    __shared__ float dinv[DT];
    __shared__ int   histS[LQ][BINS];

    const int b   = blockIdx.y;
    const int d0  = blockIdx.x * DT;
    const int tid = threadIdx.x;

    // ---- stage masked q embeddings (reference zeroes masked rows) ----
    for (int idx = tid; idx < LQ * EDIM; idx += 64) {
        int m = idx / EDIM, k = idx - m * EDIM;
        int tok = qtok[b * LQ + m];
        qs[m][k] = (tok > 1) ? emb[(size_t)tok * EDIM + k] : 0.0f;
    }
    // ---- stage masked doc embeddings ----
    for (int idx = tid; idx < DT * EDIM; idx += 64) {
        int n = idx / EDIM, k = idx - n * EDIM;
        int tok = dtok[b * LD + d0 + n];
        ds[n][k] = (tok > 1) ? emb[(size_t)tok * EDIM + k] : 0.0f;
    }
    for (int idx = tid; idx < LQ * BINS; idx += 64)
        (&histS[0][0])[idx] = 0;
    __syncthreads();

    // ---- inverse norms: threads 0..31 -> doc rows, 32..47 -> q rows ----
    if (tid < DT) {
        float s = 0.0f;
        for (int k = 0; k < EDIM; ++k) { float v = ds[tid][k]; s += v * v; }
        dinv[tid] = 1.0f / (sqrtf(s) + 1e-13f);
    } else if (tid - DT < LQ) {
        int m = tid - DT;
        float s = 0.0f;
        for (int k = 0; k < EDIM; ++k) { float v = qs[m][k]; s += v * v; }
        qinv[m] = 1.0f / (sqrtf(s) + 1e-13f);
    }
    __syncthreads();

    // ---- WMMA: D(16x16) += A(16x4) x B(4x16), 75 K-steps over E=300 ----
    const int wave  = tid >> 5;
    const int lane  = tid & 31;
    const int l16   = lane & 15;         // M for A, N for B
    const int khalf = (lane >> 4) << 1;  // 0 for lanes 0-15, 2 for 16-31

    const float ai = qinv[l16];
    const float bi = dinv[wave * 16 + l16];

    v8f c = {};
    for (int k0 = 0; k0 < EDIM; k0 += 4) {
        v2f a, bf;
        a.x  = qs[l16][k0 + khalf]     * ai;
        a.y  = qs[l16][k0 + khalf + 1] * ai;
        bf.x = ds[wave * 16 + l16][k0 + khalf]     * bi;
        bf.y = ds[wave * 16 + l16][k0 + khalf + 1] * bi;
        c = __builtin_amdgcn_wmma_f32_16x16x4_f32(
                /*neg_a=*/false, a, /*neg_b=*/false, bf,
                /*c_mod=*/(short)0, c, /*reuse_a=*/false, /*reuse_b=*/false);
    }

    // ---- histogram: C/D layout: VGPR r, lane l -> M = r + (l<16?0:8) ----
    const int mbase = (lane < 16) ? 0 : 8;
#pragma unroll
    for (int r = 0; r < 8; ++r) {
        float cv = c[r];
        int bin = (int)floorf((cv + 1.0f) * (BINS * 0.5f));
        bin = bin < 0 ? 0 : (bin > BINS - 1 ? BINS - 1 : bin);
        atomicAdd(&histS[mbase + r][bin], 1);
    }
    __syncthreads();

    // ---- flush non-zero LDS histogram entries to global ----
    const int* hflat = &histS[0][0];
    for (int idx = tid; idx < LQ * BINS; idx += 64) {
        int v = hflat[idx];
        if (v) atomicAdd(&ghist[b * LQ * BINS + idx], v);
    }
}

// ---------------------------------------------------------------------------
// Kernel 2: matching FFN on log1p(hist), gate FFN on raw q_emb, masked
// softmax over LQ, weighted sum -> out[b]. One block per batch.
// ---------------------------------------------------------------------------
__global__ __launch_bounds__(256) void drmm_tail(
    const int*   __restrict__ qtok,
    const int*   __restrict__ ghist,
    const float* __restrict__ emb,
    const float* __restrict__ mw1, const float* __restrict__ mb1,
    const float* __restrict__ mw2, const float* __restrict__ mb2,
    const float* __restrict__ gw1, const float* __restrict__ gb1,
    const float* __restrict__ gw2, const float* __restrict__ gb2,
    float*       __restrict__ out)
{
    __shared__ float qe[LQ][LPAD];     // raw masked q embeddings
    __shared__ float gh[LQ][EDIM];     // gate hidden
    __shared__ float hb[LQ][BINS];     // matching hidden
    __shared__ float grs[LQ], matchS[LQ], qm[LQ];

    const int b = blockIdx.x;
    const int tid = threadIdx.x;

    for (int idx = tid; idx < LQ * EDIM; idx += 256) {
        int m = idx / EDIM, k = idx - m * EDIM;
        int tok = qtok[b * LQ + m];
        qe[m][k] = (tok > 1) ? emb[(size_t)tok * EDIM + k] : 0.0f;
    }
    if (tid < LQ) qm[tid] = (qtok[b * LQ + tid] > 1) ? 1.0f : 0.0f;
    __syncthreads();

    // gate hidden: gh[q][e] = tanh(sum_k qe[q][k] * gw1[k][e] + gb1[e])
    for (int t = tid; t < LQ * EDIM; t += 256) {
        int q = t / EDIM, e = t - q * EDIM;
        float s = gb1[e];
        for (int k = 0; k < EDIM; ++k) s += qe[q][k] * gw1[k * EDIM + e];
        gh[q][e] = tanhf(s);
    }
    // matching hidden: hb[q][j] = tanh(sum_k log1p(hist[q][k]) * mw1[k][j] + mb1[j])
    for (int t = tid; t < LQ * BINS; t += 256) {
        int q = t / BINS, j = t - q * BINS;
        const int* hq = ghist + (b * LQ + q) * BINS;
        float s = mb1[j];
        for (int k = 0; k < BINS; ++k) s += log1pf((float)hq[k]) * mw1[k * BINS + j];
        hb[q][j] = tanhf(s);
    }
    __syncthreads();

    if (tid < LQ) {
        float s = gb2[0];
        for (int e = 0; e < EDIM; ++e) s += gh[tid][e] * gw2[e];
        grs[tid] = tanhf(s);
        float s2 = mb2[0];
        for (int j = 0; j < BINS; ++j) s2 += hb[tid][j] * mw2[j];
        matchS[tid] = tanhf(s2);
    }
    __syncthreads();

    if (tid == 0) {
        float xmax = -INFINITY;
        for (int q = 0; q < LQ; ++q)
            if (qm[q] > 0.0f && grs[q] > xmax) xmax = grs[q];
        float ssum = 0.0f, acc = 0.0f;
        for (int q = 0; q < LQ; ++q) {
            float e = (qm[q] > 0.0f) ? expf(grs[q] - xmax) : 0.0f;
            ssum += e;
            acc  += matchS[q] * e;
        }
        out[b] = (ssum > 0.0f) ? (acc / ssum) : 0.0f;
    }
}

// ---------------------------------------------------------------------------
extern "C" void kernel_launch(void* const* d_in, const int* in_sizes, int n_in,
                              void* d_out, int out_size, void* d_ws, size_t ws_size,
                              hipStream_t stream) {
    const int*   qtok = (const int*)  d_in[0];
    const int*   dtok = (const int*)  d_in[1];
    const float* emb  = (const float*)d_in[2];
    const float* mw1  = (const float*)d_in[3];
    const float* mb1  = (const float*)d_in[4];
    const float* mw2  = (const float*)d_in[5];
    const float* mb2  = (const float*)d_in[6];
    const float* gw1  = (const float*)d_in[7];
    const float* gb1  = (const float*)d_in[8];
    const float* gw2  = (const float*)d_in[9];
    const float* gb2  = (const float*)d_in[10];
    float* out  = (float*)d_out;
    int*   hist = (int*)d_ws;                 // B*LQ*BINS ints = 120 KiB

    const int nhist = BB * LQ * BINS;
    drmm_zero_hist<<<(nhist + 255) / 256, 256, 0, stream>>>(hist, nhist);

    dim3 grid(LD / DT, BB);
    drmm_cos_hist<<<grid, 64, 0, stream>>>(qtok, dtok, emb, hist);

    drmm_tail<<<BB, 256, 0, stream>>>(qtok, hist, emb,
                                      mw1, mb1, mw2, mb2,
                                      gw1, gb1, gw2, gb2, out);
}